// LaplaceLayer_15951508537781
// MI455X (gfx1250) — compile-verified
//
#include <hip/hip_runtime.h>
#include <hip/hip_bf16.h>

// B=8, C_IN=32, C_OUT=32, P=16, N=2048
#define BB  8
#define CI  32
#define CO  32
#define PP  16
#define NN  2048
#define TWO_PI 6.283185307179586f

typedef __attribute__((ext_vector_type(8))) float v8f;
typedef __attribute__((ext_vector_type(2))) float v2f;

// ---------------- LDS radix-2 FFT (N=2048, 256 threads, 11 stages) ----------
__device__ __forceinline__ void fft_lds(float2* s, float sign) {
  for (int len = 2; len <= NN; len <<= 1) {
    const int hl = len >> 1;
    const float ang = sign * TWO_PI / (float)len;
    for (int k = threadIdx.x; k < (NN / 2); k += 256) {
      const int grp = k / hl;
      const int j   = k - grp * hl;
      const int i1  = grp * len + j;
      const int i2  = i1 + hl;
      float sn, cs;
      __sincosf(ang * (float)j, &sn, &cs);
      const float2 v = s[i2];
      const float2 u = s[i1];
      const float tr = cs * v.x - sn * v.y;
      const float ti = cs * v.y + sn * v.x;
      s[i1] = make_float2(u.x + tr, u.y + ti);
      s[i2] = make_float2(u.x - tr, u.y - ti);
    }
    __syncthreads();
  }
}

// K1: alpha[b][i][n] = FFT_n(x[b][i][:]).  grid = B*CI rows.
__global__ void k_fft_fwd(const float* __restrict__ x, float2* __restrict__ alpha) {
  __shared__ float2 s[NN];
  const int row = blockIdx.x;
  const float* xr = x + (size_t)row * NN;
  for (int idx = threadIdx.x; idx < NN; idx += 256) {
    const int r = __brev((unsigned)idx) >> 21;  // 11-bit reversal
    s[r] = make_float2(xr[idx], 0.0f);
  }
  __syncthreads();
  fft_lds(s, -1.0f);
  float2* out = alpha + (size_t)row * NN;
  for (int idx = threadIdx.x; idx < NN; idx += 256) out[idx] = s[idx];
}

// K2: r2[b][o][p] = -sum_{i,n} alpha[b,i,n] * res/(i*w_n - pole).  grid = CO*PP.
__global__ void k_resid2(const float2* __restrict__ alpha,
                         const float* __restrict__ pre, const float* __restrict__ pim,
                         const float* __restrict__ rre, const float* __restrict__ rim,
                         float2* __restrict__ r2) {
  const int op = blockIdx.x;            // o*16+p
  float aRe[BB], aIm[BB];
#pragma unroll
  for (int b = 0; b < BB; b++) { aRe[b] = 0.0f; aIm[b] = 0.0f; }

  for (int i = 0; i < CI; i++) {
    const int w = i * (CO * PP) + op;   // (i*CO + o)*PP + p
    const float pr = pre[w], pi2 = pim[w];
    const float rr = rre[w], ri = rim[w];
    for (int n = threadIdx.x; n < NN; n += 256) {
      const float fr = (n < NN / 2) ? (float)n : (float)(n - NN);
      const float om = TWO_PI * fr;
      const float dr = -pr, di = om - pi2;
      const float inv = 1.0f / (dr * dr + di * di);
      const float hr = (rr * dr + ri * di) * inv;   // Re(res/(iw - pole))
      const float hi = (ri * dr - rr * di) * inv;   // Im(...)
      const float2* ap = alpha + (size_t)i * NN + n;
#pragma unroll
      for (int b = 0; b < BB; b++) {
        const float2 a = ap[(size_t)b * (CI * NN)];
        aRe[b] += a.x * hr - a.y * hi;
        aIm[b] += a.x * hi + a.y * hr;
      }
    }
  }

  __shared__ float2 red[256];
  for (int b = 0; b < BB; b++) {
    red[threadIdx.x] = make_float2(aRe[b], aIm[b]);
    __syncthreads();
    for (int s2 = 128; s2 > 0; s2 >>= 1) {
      if (threadIdx.x < (unsigned)s2) {
        red[threadIdx.x].x += red[threadIdx.x + s2].x;
        red[threadIdx.x].y += red[threadIdx.x + s2].y;
      }
      __syncthreads();
    }
    if (threadIdx.x == 0)
      r2[(size_t)b * (CO * PP) + op] = make_float2(-red[0].x, -red[0].y);
    __syncthreads();
  }
}

// K3: E[o][p][n] = sum_i exp(pole[i,o,p] * t_n),  t_n = n/(N-1).  grid = CO*PP.
__global__ void k_expE(const float* __restrict__ pre, const float* __restrict__ pim,
                       float2* __restrict__ E) {
  const int op = blockIdx.x;
  float er[8], ei[8];
#pragma unroll
  for (int j = 0; j < 8; j++) { er[j] = 0.0f; ei[j] = 0.0f; }
  for (int i = 0; i < CI; i++) {
    const int w = i * (CO * PP) + op;
    const float pr = pre[w], pi2 = pim[w];
#pragma unroll
    for (int j = 0; j < 8; j++) {
      const int n = threadIdx.x + 256 * j;
      const float t = (float)n * (1.0f / (float)(NN - 1));
      const float m = __expf(pr * t);
      float sn, cs;
      __sincosf(pi2 * t, &sn, &cs);
      er[j] += m * cs;
      ei[j] += m * sn;
    }
  }
#pragma unroll
  for (int j = 0; j < 8; j++) {
    const int n = threadIdx.x + 256 * j;
    E[(size_t)op * NN + n] = make_float2(er[j], ei[j]);
  }
}

// K4: Yhat[b][o][n] = sum_i alpha[b,i,n] * G[n,i,o],  G = sum_p res/(iw-pole).
// grid = NN blocks; 256 threads = (b,o) pairs exactly.
__global__ void k_resid1(const float2* __restrict__ alpha,
                         const float* __restrict__ pre, const float* __restrict__ pim,
                         const float* __restrict__ rre, const float* __restrict__ rim,
                         float2* __restrict__ Yhat) {
  __shared__ float2 sG[CI * CO];   // [i][o]
  __shared__ float2 sA[BB * CI];   // [b][i]
  const int n = blockIdx.x;
  const float fr = (n < NN / 2) ? (float)n : (float)(n - NN);
  const float om = TWO_PI * fr;

  for (int q = threadIdx.x; q < CI * CO; q += 256) {   // q = i*CO + o
    float gr = 0.0f, gi = 0.0f;
#pragma unroll
    for (int p = 0; p < PP; p++) {
      const int w = q * PP + p;
      const float pr = pre[w], pi2 = pim[w];
      const float rr = rre[w], ri = rim[w];
      const float dr = -pr, di = om - pi2;
      const float inv = 1.0f / (dr * dr + di * di);
      gr += (rr * dr + ri * di) * inv;
      gi += (ri * dr - rr * di) * inv;
    }
    sG[q] = make_float2(gr, gi);
  }
  {
    const int b = threadIdx.x >> 5, i = threadIdx.x & 31;
    sA[threadIdx.x] = alpha[((size_t)b * CI + i) * NN + n];
  }
  __syncthreads();

  const int b = threadIdx.x >> 5, o = threadIdx.x & 31;
  float yr = 0.0f, yi = 0.0f;
#pragma unroll 8
  for (int i = 0; i < CI; i++) {
    const float2 a = sA[b * CI + i];
    const float2 g = sG[i * CO + o];
    yr += a.x * g.x - a.y * g.y;
    yi += a.x * g.y + a.y * g.x;
  }
  Yhat[((size_t)b * CO + o) * NN + n] = make_float2(yr, yi);
}

// K5: out[b][o][n] = Re(IFFT(Yhat[b][o][:]))[n]  (x1 term).  grid = B*CO rows.
// The bit-reversed gather into LDS is a pure 8-byte copy -> use the CDNA5
// async global->LDS DMA path (ASYNCcnt).  `s` is the only LDS object in this
// kernel, so its allocation offset is 0 and rev*8 is the LDS byte address.
__global__ void k_ifft(const float2* __restrict__ Yhat, float* __restrict__ out) {
  __shared__ float2 s[NN];
  const int row = blockIdx.x;
  const float2* yr = Yhat + (size_t)row * NN;
  for (int idx = threadIdx.x; idx < NN; idx += 256) {
    const unsigned r = __brev((unsigned)idx) >> 21;
    const unsigned ldsoff = r * 8u;            // byte offset of s[r]
    const float2* g = yr + idx;
    asm volatile("global_load_async_to_lds_b64 %0, %1, off"
                 :: "v"(ldsoff), "v"(g) : "memory");
  }
  asm volatile("s_wait_asynccnt 0" ::: "memory");
  __syncthreads();
  fft_lds(s, +1.0f);
  for (int idx = threadIdx.x; idx < NN; idx += 256)
    out[(size_t)row * NN + idx] = s[idx].x * (1.0f / (float)NN);
}

// K6: out = WMMA_skip + (out_prev(x1) + bias + x2) via the WMMA C-operand
// chain.  One wave = 16(n) x 16(o) tile for one b.  2048 waves total ->
// 256 blocks x 256 threads (8 waves).  No live accumulator besides `c`.
__global__ void k_skip_x2(const float* __restrict__ x, const float* __restrict__ Km,
                          const float* __restrict__ bias,
                          const float2* __restrict__ r2, const float2* __restrict__ E,
                          float* __restrict__ out) {
  const int wave = (blockIdx.x * blockDim.x + threadIdx.x) >> 5;
  const int lane = threadIdx.x & 31;
  const int nt = wave & 127;        // n tile 0..127
  const int bt = wave >> 7;         // 0..15 = b*2 + ot
  const int b  = bt >> 1;
  const int o0 = (bt & 1) << 4;
  const int n0 = nt << 4;
  const int lo = lane & 15;
  const int hiHalf = lane >> 4;     // 0: rows/K-low half, 1: high half

  const int o  = o0 + lo;
  const int nb = n0 + hiHalf * 8;   // D layout: lanes>=16 hold rows M=8..15
  float4* op4 = (float4*)(out + ((size_t)b * CO + o) * NN + nb);  // 32B aligned
  const float bs = bias[o];

  // c = x1 (already in out) + bias
  v8f c;
  {
    const float4 u0 = op4[0], u1 = op4[1];
    c[0] = u0.x + bs; c[1] = u0.y + bs; c[2] = u0.z + bs; c[3] = u0.w + bs;
    c[4] = u1.x + bs; c[5] = u1.y + bs; c[6] = u1.z + bs; c[7] = u1.w + bs;
  }

  // c += x2 = Re( sum_p r2[b,o,p]*E[o,p,n] ) / N   (r2 pre-scaled by 1/N)
  const float invN = 1.0f / (float)NN;
  const float2* r2p = r2 + ((size_t)b * CO + o) * PP;
  const float4* Ep4 = (const float4*)(E + (size_t)(o * PP) * NN + nb);
#pragma unroll 4
  for (int p = 0; p < PP; p++) {
    const float2 rv = r2p[p];
    const float rx = rv.x * invN, ry = rv.y * invN;
    const float4* e4 = Ep4 + (size_t)p * (NN / 2);
#pragma unroll
    for (int q = 0; q < 4; q++) {
      const float4 ev = e4[q];
      c[2 * q]     += rx * ev.x - ry * ev.y;
      c[2 * q + 1] += rx * ev.z - ry * ev.w;
    }
  }

  // skip GEMM on top of c: fp32 A 16x4 (lanes 0-15: K=k0,k0+1; 16-31: K=k0+2,3)
#pragma unroll
  for (int ks = 0; ks < 8; ks++) {
    const int i0 = ks * 4 + hiHalf * 2;
    v2f a, bv;
    const float* xb = x + ((size_t)b * CI + i0) * NN + n0 + lo;
    a.x = xb[0];
    a.y = xb[NN];
    bv.x = Km[i0 * CO + o0 + lo];
    bv.y = Km[(i0 + 1) * CO + o0 + lo];
    c = __builtin_amdgcn_wmma_f32_16x16x4_f32(false, a, false, bv, (short)0, c,
                                              false, false);
  }

  op4[0] = make_float4(c[0], c[1], c[2], c[3]);
  op4[1] = make_float4(c[4], c[5], c[6], c[7]);
}

extern "C" void kernel_launch(void* const* d_in, const int* in_sizes, int n_in,
                              void* d_out, int out_size, void* d_ws, size_t ws_size,
                              hipStream_t stream) {
  (void)in_sizes; (void)n_in; (void)out_size; (void)ws_size;
  const float* x    = (const float*)d_in[0];
  const float* wpr  = (const float*)d_in[1];
  const float* wpi  = (const float*)d_in[2];
  const float* wrr  = (const float*)d_in[3];
  const float* wri  = (const float*)d_in[4];
  const float* kmat = (const float*)d_in[5];
  const float* bias = (const float*)d_in[6];
  float* out = (float*)d_out;

  // Workspace layout (~16.1 MB total):
  //   [0,4MB)   alpha  (B*CI*N complex)
  //   [4,8MB)   Yhat   (B*CO*N complex)
  //   [8,16MB)  E      (CO*P*N complex)
  //   [16MB,+)  r2     (B*CO*P complex, 32 KB)
  char* ws = (char*)d_ws;
  float2* alpha = (float2*)(ws);
  float2* Yhat  = (float2*)(ws + ((size_t)4 << 20));
  float2* E     = (float2*)(ws + ((size_t)8 << 20));
  float2* r2    = (float2*)(ws + ((size_t)16 << 20));

  k_fft_fwd<<<BB * CI, 256, 0, stream>>>(x, alpha);
  k_resid2 <<<CO * PP, 256, 0, stream>>>(alpha, wpr, wpi, wrr, wri, r2);
  k_expE   <<<CO * PP, 256, 0, stream>>>(wpr, wpi, E);
  k_resid1 <<<NN,      256, 0, stream>>>(alpha, wpr, wpi, wrr, wri, Yhat);
  k_ifft   <<<BB * CO, 256, 0, stream>>>(Yhat, out);
  k_skip_x2<<<(BB * 2 * (NN / 16)) / 8, 256, 0, stream>>>(x, kmat, bias, r2, E, out);
}